// Megnet_NodeModel_53970559042217
// MI455X (gfx1250) — compile-verified
//
#include <hip/hip_runtime.h>
#include <hip/hip_bf16.h>

// CDNA5 / gfx1250 wave32 WMMA types
typedef float v2f __attribute__((ext_vector_type(2)));
typedef float v8f __attribute__((ext_vector_type(8)));

#define NDIM 64
#define KDIM 192   // 3*NDIM

// ---------------------------------------------------------------------------
// Phase 1: scatter-add of edge_attr into sums[N,64] + edge counts.
// One thread per (edge, 4-dim group): float4 coalesced read, 4 f32 atomics.
// ---------------------------------------------------------------------------
__global__ __launch_bounds__(256) void megnet_scatter(
    const float* __restrict__ edge_attr,   // [E,64]
    const int*   __restrict__ src,         // edge_index row 0, [E]
    float* __restrict__ sums,              // [N,64]
    float* __restrict__ counts,            // [N]
    int E)
{
    int id = blockIdx.x * blockDim.x + threadIdx.x;
    if (id >= E * 16) return;
    int e = id >> 4;
    int q = id & 15;                       // which float4 of the 64-dim row
    int s = src[e];
    const float4 v = ((const float4*)edge_attr)[(size_t)e * 16 + q];
    float* dst = sums + (size_t)s * NDIM + q * 4;
    atomicAdd(dst + 0, v.x);
    atomicAdd(dst + 1, v.y);
    atomicAdd(dst + 2, v.z);
    atomicAdd(dst + 3, v.w);
    if (q == 0) atomicAdd(&counts[s], 1.0f);
}

// ---------------------------------------------------------------------------
// Phase 2: fused  out = relu([x|v_e|u[batch]] @ W1 + b1) @ W2 + b2
// 128 threads = 4 wave32; each wave owns one 16-row tile.
// W1/W2 staged in LDS per block; comb tile staged in LDS per wave;
// both GEMMs via V_WMMA_F32_16X16X4_F32 (fp32-exact).
// ---------------------------------------------------------------------------
__global__ __launch_bounds__(128) void megnet_mlp(
    const float* __restrict__ x,       // [N,64]
    const float* __restrict__ u,       // [B,64]
    const int*   __restrict__ batch,   // [N]
    const float* __restrict__ W1,      // [192,64]
    const float* __restrict__ b1,      // [64]
    const float* __restrict__ W2,      // [64,64]
    const float* __restrict__ b2,      // [64]
    const float* __restrict__ sums,    // [N,64]
    const float* __restrict__ counts,  // [N]
    float* __restrict__ out,           // [N,64]
    int nTiles, int N)
{
    __shared__ float sW1[KDIM * NDIM];        // 48 KB
    __shared__ float sW2[NDIM * NDIM];        // 16 KB
    __shared__ float sComb[4][16 * KDIM];     // 48 KB (per-wave tiles; reused for h)

    const int tid  = threadIdx.x;
    const int wave = tid >> 5;
    const int lane = tid & 31;
    const int half = lane >> 4;               // 0: K=k0,k0+1  1: K=k0+2,k0+3
    const int l16  = lane & 15;

    // --- stage weights (whole block, float4 coalesced) ---
    for (int i = tid; i < (KDIM * NDIM) / 4; i += blockDim.x)
        ((float4*)sW1)[i] = ((const float4*)W1)[i];
    for (int i = tid; i < (NDIM * NDIM) / 4; i += blockDim.x)
        ((float4*)sW2)[i] = ((const float4*)W2)[i];
    __syncthreads();

    const int tile = blockIdx.x * 4 + wave;
    if (tile >= nTiles) return;               // uniform per-wave exit; no syncs below
    const int row0 = tile * 16;

    float* myComb = &sComb[wave][0];          // [16][192]

    // --- stage comb tile: 16 rows x 192 cols = 768 float4 slots ---
    for (int slot = lane; slot < 768; slot += 32) {
        int r   = slot / 48;                  // tile row
        int c4  = slot % 48;                  // float4 column
        int col = c4 * 4;
        int row = row0 + r; if (row >= N) row = N - 1;   // clamp (EXEC stays full)
        float4 v;
        if (col < 64) {
            v = ((const float4*)(x + (size_t)row * NDIM))[c4];
        } else if (col < 128) {
            float scale = 1.0f / fmaxf(counts[row], 1.0f);
            float4 s4 = ((const float4*)(sums + (size_t)row * NDIM))[c4 - 16];
            v.x = s4.x * scale; v.y = s4.y * scale;
            v.z = s4.z * scale; v.w = s4.w * scale;
        } else {
            int b = batch[row];
            v = ((const float4*)(u + (size_t)b * NDIM))[c4 - 32];
        }
        ((float4*)myComb)[slot] = v;
    }
    // same-wave LDS ops are in-order; compiler inserts s_wait_dscnt for data deps

    // biases for this lane's 4 n-tiles
    float bias1[4], bias2[4];
#pragma unroll
    for (int nt = 0; nt < 4; ++nt) {
        bias1[nt] = b1[nt * 16 + l16];
        bias2[nt] = b2[nt * 16 + l16];
    }

    // --- layer 1: [16,192] @ [192,64], 48 k-steps x 4 n-tiles of 16x16x4 f32 WMMA
    v8f acc[4] = {};
    for (int k0 = 0; k0 < KDIM; k0 += 4) {
        const int ka = k0 + half * 2;
        v2f a;
        a.x = myComb[l16 * KDIM + ka];
        a.y = myComb[l16 * KDIM + ka + 1];
#pragma unroll
        for (int nt = 0; nt < 4; ++nt) {
            const int n = nt * 16 + l16;
            v2f b;
            b.x = sW1[ka * NDIM + n];
            b.y = sW1[(ka + 1) * NDIM + n];
            acc[nt] = __builtin_amdgcn_wmma_f32_16x16x4_f32(
                false, a, false, b, (short)0, acc[nt], false, false);
        }
    }

    // --- relu + b1, transpose C/D layout -> A layout through LDS (reuse comb)
    float* hT = myComb;                        // [16][64]
#pragma unroll
    for (int nt = 0; nt < 4; ++nt) {
        const int n = nt * 16 + l16;
#pragma unroll
        for (int r = 0; r < 8; ++r) {
            const int m = r + half * 8;
            hT[m * NDIM + n] = fmaxf(acc[nt][r] + bias1[nt], 0.0f);
        }
    }

    // --- layer 2: [16,64] @ [64,64]
    v8f acc2[4] = {};
    for (int k0 = 0; k0 < NDIM; k0 += 4) {
        const int ka = k0 + half * 2;
        v2f a;
        a.x = hT[l16 * NDIM + ka];
        a.y = hT[l16 * NDIM + ka + 1];
#pragma unroll
        for (int nt = 0; nt < 4; ++nt) {
            const int n = nt * 16 + l16;
            v2f b;
            b.x = sW2[ka * NDIM + n];
            b.y = sW2[(ka + 1) * NDIM + n];
            acc2[nt] = __builtin_amdgcn_wmma_f32_16x16x4_f32(
                false, a, false, b, (short)0, acc2[nt], false, false);
        }
    }

    // --- + b2, store (lanes 0-15 and 16-31 each write 16 contiguous floats)
#pragma unroll
    for (int nt = 0; nt < 4; ++nt) {
        const int n = nt * 16 + l16;
#pragma unroll
        for (int r = 0; r < 8; ++r) {
            const int m = r + half * 8;
            const int row = row0 + m;
            if (row < N)
                out[(size_t)row * NDIM + n] = acc2[nt][r] + bias2[nt];
        }
    }
}

// ---------------------------------------------------------------------------
extern "C" void kernel_launch(void* const* d_in, const int* in_sizes, int n_in,
                              void* d_out, int out_size, void* d_ws, size_t ws_size,
                              hipStream_t stream) {
    const float* x          = (const float*)d_in[0];
    const int*   edge_index = (const int*)  d_in[1];   // [2,E], row 0 = src
    const float* edge_attr  = (const float*)d_in[2];
    const float* u          = (const float*)d_in[3];
    const int*   batch      = (const int*)  d_in[4];
    const float* W1         = (const float*)d_in[5];
    const float* b1         = (const float*)d_in[6];
    const float* W2         = (const float*)d_in[7];
    const float* b2         = (const float*)d_in[8];
    float*       out        = (float*)d_out;

    const int N = in_sizes[0] / NDIM;
    const int E = in_sizes[2] / NDIM;

    float* sums   = (float*)d_ws;              // N*64 floats
    float* counts = sums + (size_t)N * NDIM;   // N floats

    // zero accumulators each call (ws is not re-poisoned between replays)
    hipMemsetAsync(d_ws, 0, (size_t)N * (NDIM + 1) * sizeof(float), stream);

    const int* src = edge_index;               // first row
    const int scatterThreads = E * 16;
    megnet_scatter<<<(scatterThreads + 255) / 256, 256, 0, stream>>>(
        edge_attr, src, sums, counts, E);

    const int nTiles = (N + 15) / 16;
    const int blocks = (nTiles + 3) / 4;       // 4 waves (tiles) per block
    megnet_mlp<<<blocks, 128, 0, stream>>>(
        x, u, batch, W1, b1, W2, b2, sums, counts, out, nTiles, N);
}